// GTN_layer_20942260536102
// MI455X (gfx1250) — compile-verified
//
#include <hip/hip_runtime.h>
#include <hip/hip_bf16.h>
#include <stddef.h>

// ---------------------------------------------------------------------------
// GTN layer for MI455X (gfx1250, wave32, WMMA). D=256, H=8, HD=32.
// ---------------------------------------------------------------------------

typedef float v2f __attribute__((ext_vector_type(2)));
typedef float v8f __attribute__((ext_vector_type(8)));

#define DMODEL 256
#define NHEAD  8
#define HDIM   32
#define TM     128   // M tile per workgroup
#define KC     64    // K chunk staged in LDS
#define LDA    68    // As row stride: 68 % 64 == 4 -> conflict-free v2f frag reads
#define LDW    68    // Wt (transposed W) row stride, same property
#define LDAOLD 66    // row stride for LN kernel's A tile (legacy layout)
#define NEG_INF (-1e30f)
#define SCALE   0.17677669529663688f   // 1/sqrt(32)

// ---------------------------------------------------------------------------
// fp32 WMMA GEMM:  C[M x 256] = A[M x 256] @ W[256 x 256] + bias
// 256 threads / 8 waves. Wave tile = 64x64 (4x4 WMMA tiles): per K-step only
// 4 A-frags + 4 B-frags (b64 each) feed 16 back-to-back v_wmma_f32_16x16x4_f32.
// W is transposed on the fly into LDS so B-frags are single ds_load_b64.
// ---------------------------------------------------------------------------
__global__ __launch_bounds__(256) void gemm256_kernel(
    const float* __restrict__ A, const float* __restrict__ W,
    const float* __restrict__ bias, float* __restrict__ C, int M)
{
    __shared__ float As[TM * LDA];       // 128 x 64 (+pad)      ~34 KB
    __shared__ float Wt[DMODEL * LDW];   // 256 cols x 64 k (+pad) ~68 KB

    const int tid  = threadIdx.x;
    const int m0   = blockIdx.x * TM;
    const int wv   = tid >> 5;           // wave id 0..7
    const int lane = tid & 31;
    const int hf   = lane >> 4;          // half-wave: K sub-offset
    const int lr   = lane & 15;
    const int ms   = (wv & 1) * 64;      // wave M offset (4 strips of 16)
    const int nw   = (wv >> 1) * 64;     // wave N offset (4 tiles of 16)

    v8f acc[16];
    const v8f z = {0.f,0.f,0.f,0.f,0.f,0.f,0.f,0.f};
#pragma unroll
    for (int t = 0; t < 16; ++t) acc[t] = z;

    for (int kc = 0; kc < DMODEL; kc += KC) {
        // ---- stage A tile: 128 x 64 floats (2048 float4 slots, 8/thread)
#pragma unroll
        for (int i = 0; i < 8; ++i) {
            int slot = tid + i * 256;
            int row  = slot >> 4;
            int c4   = (slot & 15) << 2;
            *(float4*)(&As[row * LDA + c4]) =
                *(const float4*)(A + (size_t)(m0 + row) * DMODEL + kc + c4);
        }
        // ---- stage W transposed: thread owns column c = tid, all 32 k-pairs.
        // Global reads: two coalesced 128B rows; LDS writes: b64, <=2-way conflict.
#pragma unroll 8
        for (int i = 0; i < 32; ++i) {
            int k = 2 * i;
            v2f p;
            p.x = W[(size_t)(kc + k    ) * DMODEL + tid];
            p.y = W[(size_t)(kc + k + 1) * DMODEL + tid];
            *(v2f*)(&Wt[tid * LDW + k]) = p;
        }
        __syncthreads();

        // ---- compute: 16 K-steps of 4; 8 b64 frag loads -> 16 WMMA burst
#pragma unroll 2
        for (int kk = 0; kk < KC; kk += 4) {
            v2f af[4], bf[4];
#pragma unroll
            for (int i = 0; i < 4; ++i)
                af[i] = *(const v2f*)(&As[(ms + i * 16 + lr) * LDA + kk + 2 * hf]);
#pragma unroll
            for (int j = 0; j < 4; ++j)
                bf[j] = *(const v2f*)(&Wt[(nw + j * 16 + lr) * LDW + kk + 2 * hf]);
#pragma unroll
            for (int i = 0; i < 4; ++i)
#pragma unroll
                for (int j = 0; j < 4; ++j)
                    acc[i * 4 + j] = __builtin_amdgcn_wmma_f32_16x16x4_f32(
                        false, af[i], false, bf[j], (short)0, acc[i * 4 + j],
                        false, false);
        }
        __syncthreads();
    }

    // ---- epilogue: bias + store (VGPR r: lanes0-15 row +r, lanes16-31 row +r+8)
#pragma unroll
    for (int j = 0; j < 4; ++j) {
        int col = nw + j * 16 + lr;
        float bv = bias[col];
#pragma unroll
        for (int i = 0; i < 4; ++i) {
#pragma unroll
            for (int r = 0; r < 8; ++r) {
                int row = m0 + ms + i * 16 + r + hf * 8;
                C[(size_t)row * DMODEL + col] = acc[i * 4 + j][r] + bv;
            }
        }
    }
}

// ---------------------------------------------------------------------------
// Final projection:  out = LayerNorm( attn @ Wo + bo + Xp ) * g + b
// 1 wave = 16 full rows so LN reduces with 4 __shfl_xor steps per 16-lane half.
// ---------------------------------------------------------------------------
__global__ __launch_bounds__(256) void gemm256_ln_kernel(
    const float* __restrict__ A, const float* __restrict__ W,
    const float* __restrict__ bias, const float* __restrict__ Xp,
    const float* __restrict__ g, const float* __restrict__ bln,
    float* __restrict__ out, int M)
{
    __shared__ float As[TM * LDAOLD];
    __shared__ float Ws[KC * DMODEL];

    const int tid  = threadIdx.x;
    const int m0   = blockIdx.x * TM;
    const int wv   = tid >> 5;
    const int lane = tid & 31;
    const int hf   = lane >> 4;
    const int lr   = lane & 15;
    const int mw   = wv * 16;

    v8f acc[16];
    const v8f z = {0.f,0.f,0.f,0.f,0.f,0.f,0.f,0.f};
#pragma unroll
    for (int t = 0; t < 16; ++t) acc[t] = z;

    for (int kc = 0; kc < DMODEL; kc += KC) {
#pragma unroll
        for (int i = 0; i < 8; ++i) {
            int slot = tid + i * 256;
            int row  = slot >> 4;
            int c4   = (slot & 15) << 2;
            float4 v = *(const float4*)(A + (size_t)(m0 + row) * DMODEL + kc + c4);
            float* dst = &As[row * LDAOLD + c4];
            dst[0] = v.x; dst[1] = v.y; dst[2] = v.z; dst[3] = v.w;
        }
#pragma unroll
        for (int i = 0; i < 16; ++i) {
            int slot = tid + i * 256;
            int row  = slot >> 6;
            int c4   = (slot & 63) << 2;
            *(float4*)(&Ws[row * DMODEL + c4]) =
                *(const float4*)(W + (size_t)(kc + row) * DMODEL + c4);
        }
        __syncthreads();
#pragma unroll 4
        for (int kk = 0; kk < KC; kk += 4) {
            v2f a = *(const v2f*)(&As[(mw + lr) * LDAOLD + kk + hf * 2]);
#pragma unroll
            for (int t = 0; t < 16; ++t) {
                int col = t * 16 + lr;
                v2f b;
                b.x = Ws[(kk + hf * 2    ) * DMODEL + col];
                b.y = Ws[(kk + hf * 2 + 1) * DMODEL + col];
                acc[t] = __builtin_amdgcn_wmma_f32_16x16x4_f32(
                    false, a, false, b, (short)0, acc[t], false, false);
            }
        }
        __syncthreads();
    }

    // h = acc + bo + Xp
#pragma unroll
    for (int t = 0; t < 16; ++t) {
        int col = t * 16 + lr;
        float bv = bias[col];
#pragma unroll
        for (int r = 0; r < 8; ++r) {
            int row = m0 + mw + r + hf * 8;
            acc[t][r] += bv + Xp[(size_t)row * DMODEL + col];
        }
    }

    // row stats
    float s1[8], s2[8];
#pragma unroll
    for (int r = 0; r < 8; ++r) { s1[r] = 0.f; s2[r] = 0.f; }
#pragma unroll
    for (int t = 0; t < 16; ++t)
#pragma unroll
        for (int r = 0; r < 8; ++r) {
            float h = acc[t][r];
            s1[r] += h; s2[r] += h * h;
        }
#pragma unroll
    for (int r = 0; r < 8; ++r) {
#pragma unroll
        for (int mask = 8; mask >= 1; mask >>= 1) {
            s1[r] += __shfl_xor(s1[r], mask, 16);
            s2[r] += __shfl_xor(s2[r], mask, 16);
        }
    }
    float mu[8], rs[8];
#pragma unroll
    for (int r = 0; r < 8; ++r) {
        mu[r] = s1[r] * (1.0f / DMODEL);
        float var = s2[r] * (1.0f / DMODEL) - mu[r] * mu[r];
        rs[r] = rsqrtf(var + 1e-5f);
    }
#pragma unroll
    for (int t = 0; t < 16; ++t) {
        int col = t * 16 + lr;
        float gv = g[col], bb = bln[col];
#pragma unroll
        for (int r = 0; r < 8; ++r) {
            int row = m0 + mw + r + hf * 8;
            out[(size_t)row * DMODEL + col] = (acc[t][r] - mu[r]) * rs[r] * gv + bb;
        }
    }
}

// ---------------------------------------------------------------------------
// Attention stage kernels
// ---------------------------------------------------------------------------
__device__ __forceinline__ void atomicMaxF(float* addr, float val) {
    if (val >= 0.f) atomicMax((int*)addr, __float_as_int(val));
    else            atomicMin((unsigned int*)addr, __float_as_uint(val));
}

// m[i,h] = selfS[i,h] = (Q[i,h] . K[i,h]) * scale      (N*H threads)
__global__ __launch_bounds__(256) void self_scores_kernel(
    const float* __restrict__ Q, const float* __restrict__ K,
    float* __restrict__ m, float* __restrict__ selfS, int total)
{
    int idx = blockIdx.x * 256 + threadIdx.x;
    if (idx >= total) return;
    int i = idx >> 3, h = idx & 7;
    const float4* q = (const float4*)(Q + (size_t)i * DMODEL + h * HDIM);
    const float4* k = (const float4*)(K + (size_t)i * DMODEL + h * HDIM);
    float s = 0.f;
#pragma unroll
    for (int j = 0; j < 8; ++j) {
        float4 a = q[j], b = k[j];
        s += a.x * b.x + a.y * b.y + a.z * b.z + a.w * b.w;
    }
    s *= SCALE;
    m[idx] = s;
    selfS[idx] = s;
}

// per-(edge,head) score + running max into m[src,h]       (E*H threads)
__global__ __launch_bounds__(256) void edge_scores_kernel(
    const float* __restrict__ Q, const float* __restrict__ K,
    const float* __restrict__ Ef, const int* __restrict__ src,
    const int* __restrict__ tgt, float* __restrict__ score,
    float* __restrict__ m, int total)
{
    int idx = blockIdx.x * 256 + threadIdx.x;
    if (idx >= total) return;
    int e = idx >> 3, h = idx & 7;
    int s = src[e], t = tgt[e];
    if (s == t) { score[idx] = NEG_INF; return; }   // diagonal overridden by self
    const float4* q  = (const float4*)(Q  + (size_t)s * DMODEL + h * HDIM);
    const float4* k  = (const float4*)(K  + (size_t)t * DMODEL + h * HDIM);
    const float4* ef = (const float4*)(Ef + (size_t)e * DMODEL + h * HDIM);
    float acc = 0.f;
#pragma unroll
    for (int j = 0; j < 8; ++j) {
        float4 a = q[j], b = k[j], c = ef[j];
        acc += a.x*b.x + a.y*b.y + a.z*b.z + a.w*b.w;   // Q.K
        acc += a.x*c.x + a.y*c.y + a.z*c.z + a.w*c.w;   // Q.E
        acc += c.x*b.x + c.y*b.y + c.z*b.z + c.w*b.w;   // E.K
    }
    float sc = acc * SCALE;
    score[idx] = sc;
    atomicMaxF(&m[s * NHEAD + h], sc);
}

// ssum = exp(self - m); acc[i,h,:] = w0 * V[i,h,:]        (N*H threads)
__global__ __launch_bounds__(256) void init_acc_kernel(
    const float* __restrict__ V, const float* __restrict__ m,
    const float* __restrict__ selfS, float* __restrict__ ssum,
    float* __restrict__ acc, int total)
{
    int idx = blockIdx.x * 256 + threadIdx.x;
    if (idx >= total) return;
    int i = idx >> 3, h = idx & 7;
    float w0 = __expf(selfS[idx] - m[idx]);
    ssum[idx] = w0;
    const float4* v = (const float4*)(V + (size_t)i * DMODEL + h * HDIM);
    float4* a = (float4*)(acc + (size_t)i * DMODEL + h * HDIM);
#pragma unroll
    for (int j = 0; j < 8; ++j) {
        float4 t = v[j];
        a[j] = make_float4(w0 * t.x, w0 * t.y, w0 * t.z, w0 * t.w);
    }
}

// one 32-lane group per edge: coalesced atomic scatter     (E*32 threads)
// lane d adds w*V[tgt,h,d] to acc[src,h,d] for h=0..7; lanes 0..7 update ssum
__global__ __launch_bounds__(256) void edge_accum_kernel(
    const float* __restrict__ V, const float* __restrict__ m,
    const float* __restrict__ score, const int* __restrict__ src,
    const int* __restrict__ tgt, float* __restrict__ ssum,
    float* __restrict__ acc, int total)
{
    int idx = blockIdx.x * 256 + threadIdx.x;
    if (idx >= total) return;
    int e = idx >> 5, d = idx & 31;
    int s = src[e], t = tgt[e];
    const float* sc = score + (size_t)e * NHEAD;
    const float* mv = m + (size_t)s * NHEAD;
    if (d < NHEAD) {
        float w = __expf(sc[d] - mv[d]);
        if (w > 0.f) atomicAdd(&ssum[s * NHEAD + d], w);
    }
    const float* vp = V + (size_t)t * DMODEL + d;
    float* ap = acc + (size_t)s * DMODEL + d;
#pragma unroll
    for (int h = 0; h < NHEAD; ++h) {
        float w = __expf(sc[h] - mv[h]);       // NEG_INF path underflows to 0
        if (w > 0.f) atomicAdd(&ap[h * HDIM], w * vp[h * HDIM]);
    }
}

// acc /= ssum                                             (N*256 threads)
__global__ __launch_bounds__(256) void normalize_kernel(
    const float* __restrict__ ssum, float* __restrict__ acc, int total)
{
    int idx = blockIdx.x * 256 + threadIdx.x;
    if (idx >= total) return;
    int i = idx >> 8, col = idx & 255, h = col >> 5;
    acc[idx] /= ssum[i * NHEAD + h];
}

// ---------------------------------------------------------------------------
extern "C" void kernel_launch(void* const* d_in, const int* in_sizes, int n_in,
                              void* d_out, int out_size, void* d_ws, size_t ws_size,
                              hipStream_t stream)
{
    const float* x    = (const float*)d_in[0];
    const int*   ei   = (const int*)  d_in[1];
    const float* ea   = (const float*)d_in[2];
    const float* Wq   = (const float*)d_in[3];
    const float* bq   = (const float*)d_in[4];
    const float* Wk   = (const float*)d_in[5];
    const float* bk   = (const float*)d_in[6];
    const float* Wv   = (const float*)d_in[7];
    const float* bv   = (const float*)d_in[8];
    const float* We   = (const float*)d_in[9];
    const float* be   = (const float*)d_in[10];
    const float* Wo   = (const float*)d_in[11];
    const float* bo   = (const float*)d_in[12];
    const float* Wi   = (const float*)d_in[13];
    const float* bi   = (const float*)d_in[14];
    const float* ln_g = (const float*)d_in[15];
    const float* ln_b = (const float*)d_in[16];

    const int N = in_sizes[0] / DMODEL;     // 4096
    const int E = in_sizes[1] / 2;          // 131072
    const int* src = ei;
    const int* tgt = ei + E;

    float* node_out = (float*)d_out;                       // (N,256)
    float* edge_out = (float*)d_out + (size_t)N * DMODEL;  // (E,256) == Efeat

    // workspace layout (floats)
    float* ws = (float*)d_ws;
    size_t off = 0;
    float* Q     = ws + off; off += (size_t)N * DMODEL;
    float* K     = ws + off; off += (size_t)N * DMODEL;
    float* V     = ws + off; off += (size_t)N * DMODEL;
    float* Xp    = ws + off; off += (size_t)N * DMODEL;
    float* score = ws + off; off += (size_t)E * NHEAD;
    float* mbuf  = ws + off; off += (size_t)N * NHEAD;
    float* selfS = ws + off; off += (size_t)N * NHEAD;
    float* ssum  = ws + off; off += (size_t)N * NHEAD;
    float* acc   = ws + off; off += (size_t)N * DMODEL;
    (void)ws_size; (void)n_in; (void)out_size;

    dim3 blk(256);

    // 1) projections (fp32 WMMA GEMMs)
    gemm256_kernel<<<dim3(N / TM), blk, 0, stream>>>(x, Wq, bq, Q, N);
    gemm256_kernel<<<dim3(N / TM), blk, 0, stream>>>(x, Wk, bk, K, N);
    gemm256_kernel<<<dim3(N / TM), blk, 0, stream>>>(x, Wv, bv, V, N);
    gemm256_kernel<<<dim3(N / TM), blk, 0, stream>>>(x, Wi, bi, Xp, N);
    // edge projection writes edge_out directly; it doubles as Efeat
    gemm256_kernel<<<dim3(E / TM), blk, 0, stream>>>(ea, We, be, edge_out, E);

    // 2) sparse attention
    int nh = N * NHEAD, eh = E * NHEAD, nd = N * DMODEL;
    int ew = E * 32;
    self_scores_kernel<<<dim3((nh + 255) / 256), blk, 0, stream>>>(Q, K, mbuf, selfS, nh);
    edge_scores_kernel<<<dim3((eh + 255) / 256), blk, 0, stream>>>(
        Q, K, edge_out, src, tgt, score, mbuf, eh);
    init_acc_kernel<<<dim3((nh + 255) / 256), blk, 0, stream>>>(V, mbuf, selfS, ssum, acc, nh);
    edge_accum_kernel<<<dim3((ew + 255) / 256), blk, 0, stream>>>(
        V, mbuf, score, src, tgt, ssum, acc, ew);
    normalize_kernel<<<dim3((nd + 255) / 256), blk, 0, stream>>>(ssum, acc, nd);

    // 3) output projection + residual + layernorm (fused)
    gemm256_ln_kernel<<<dim3(N / TM), blk, 0, stream>>>(
        acc, Wo, bo, Xp, ln_g, ln_b, node_out, N);
}